// NSLoss_49838800503231
// MI455X (gfx1250) — compile-verified
//
#include <hip/hip_runtime.h>

typedef float v2f __attribute__((ext_vector_type(2)));
typedef float v8f __attribute__((ext_vector_type(8)));

#define DIMS 128
#define KNEG 5
#define TILE 16

__device__ __forceinline__ float log_sigmoid(float x) {
    // stable: min(x,0) - log1p(exp(-|x|))
    return fminf(x, 0.0f) - log1pf(__expf(-fabsf(x)));
}

// One wave (32 lanes) handles 16 samples. For each of 6 score sets
// (1 positive + 5 negatives) build B from the gathered weight rows and
// run 32 x V_WMMA_F32_16X16X4_F32 over D=128; diagonal of the 16x16
// result holds the per-sample dot products.
__global__ __launch_bounds__(32) void nsloss_tile_kernel(
    const float* __restrict__ embs, const float* __restrict__ weights,
    const int* __restrict__ label, const int* __restrict__ negs,
    float* __restrict__ partials)
{
    __shared__ __align__(32) float Dt[KNEG + 1][32][8];

    const int lane   = threadIdx.x;   // 0..31
    const int m      = lane & 15;     // sample-in-tile / matrix row & col
    const int khalf  = lane >> 4;     // 0: K pair {0,1}, 1: K pair {2,3}
    const int sample = blockIdx.x * TILE + m;

    // Gather row ids for this sample's 6 score sets (pos + 5 neg).
    int wrow[KNEG + 1];
    wrow[0] = label[sample];
#pragma unroll
    for (int j = 0; j < KNEG; ++j)
        wrow[j + 1] = negs[(size_t)sample * KNEG + j];

    // Kick off the random-row fetches early (global_prefetch_b8).
#pragma unroll
    for (int j = 0; j < KNEG + 1; ++j) {
        const char* p = (const char*)(weights + (size_t)wrow[j] * DIMS)
                        + khalf * 256;
        __builtin_prefetch(p, 0, 3);
        __builtin_prefetch(p + 128, 0, 3);
    }

    // A fragments: lane holds emb[sample][4*kk + 2*khalf + {0,1}].
    const float* er = embs + (size_t)sample * DIMS;
    v2f a[DIMS / 4];
#pragma unroll
    for (int kk = 0; kk < DIMS / 4; ++kk)
        a[kk] = *(const v2f*)(er + kk * 4 + khalf * 2);

#pragma unroll
    for (int j = 0; j < KNEG + 1; ++j) {
        const float* wr = weights + (size_t)wrow[j] * DIMS;
        v8f acc = {};
#pragma unroll
        for (int kk = 0; kk < DIMS / 4; ++kk) {
            v2f b = *(const v2f*)(wr + kk * 4 + khalf * 2);
            // 8 args: (neg_a, A, neg_b, B, c_mod, C, reuse_a, reuse_b)
            acc = __builtin_amdgcn_wmma_f32_16x16x4_f32(
                false, a[kk], false, b, (short)0, acc, false, false);
        }
        // Dump accumulator; value at Dt[j][lane][g] is D[M][N] with
        // M = g + (lane<16 ? 0 : 8), N = lane&15.
        *(v8f*)&Dt[j][lane][0] = acc;
    }
    __syncthreads();

    float l = 0.0f;
    if (lane < TILE) {
        // diagonal (m,m): m<8 -> [m][m]; m>=8 -> [m+16][m-8]
        const int r = (m < 8) ? m : m + 16;
        const int c = (m < 8) ? m : m - 8;
        l = log_sigmoid(Dt[0][r][c]);          // positive score
#pragma unroll
        for (int j = 1; j <= KNEG; ++j)
            l += log_sigmoid(-Dt[j][r][c]);    // noise = -W[negs]
    }
    // wave32 reduction (lanes 16..31 contribute 0)
#pragma unroll
    for (int off = 16; off > 0; off >>= 1)
        l += __shfl_down(l, off, 32);
    if (lane == 0) partials[blockIdx.x] = l;
}

// Deterministic fixed-order final reduction: out = -sum(partials)/N.
__global__ __launch_bounds__(256) void nsloss_reduce_kernel(
    const float* __restrict__ partials, float* __restrict__ out,
    int ntiles, float inv_n)
{
    __shared__ float s[256];
    float acc = 0.0f;
    for (int i = threadIdx.x; i < ntiles; i += 256)
        acc += partials[i];
    s[threadIdx.x] = acc;
    __syncthreads();
    for (int step = 128; step > 0; step >>= 1) {
        if (threadIdx.x < step) s[threadIdx.x] += s[threadIdx.x + step];
        __syncthreads();
    }
    if (threadIdx.x == 0) out[0] = -s[0] * inv_n;
}

extern "C" void kernel_launch(void* const* d_in, const int* in_sizes, int n_in,
                              void* d_out, int out_size, void* d_ws, size_t ws_size,
                              hipStream_t stream) {
    const float* embs    = (const float*)d_in[0];
    const float* weights = (const float*)d_in[1];
    const int*   label   = (const int*)d_in[2];
    const int*   negs    = (const int*)d_in[3];

    const int n      = in_sizes[2];      // 65536
    const int ntiles = n / TILE;         // 4096

    float* partials = (float*)d_ws;      // ntiles floats of scratch

    nsloss_tile_kernel<<<ntiles, 32, 0, stream>>>(embs, weights, label, negs,
                                                  partials);
    nsloss_reduce_kernel<<<1, 256, 0, stream>>>(partials, (float*)d_out,
                                                ntiles, 1.0f / (float)n);
}